// GCNet_37726992728287
// MI455X (gfx1250) — compile-verified
//
#include <hip/hip_runtime.h>

typedef float v2f __attribute__((ext_vector_type(2)));
typedef float v8f __attribute__((ext_vector_type(8)));

namespace {
constexpr int kC   = 32;    // input channels
constexpr int kH   = 128;
constexpr int kW   = 256;
constexpr int kPad = 64;    // zero pad so row[kPad + w - d] covers w-d < 0
constexpr float kLog2e = 1.44269504088896340736f;
}

// Channels [0, C): softmax over a d-constant slab is exactly uniform ->
// output is the mean disparity everywhere. Pure fill, no input reads.
__global__ void gcnet_fill_left(float* __restrict__ out,
                                const int* __restrict__ min_disp,
                                const int* __restrict__ max_disp) {
  const int dmin = min_disp[0] >> 1;
  const int dmax = max_disp[0] >> 1;
  const float val = (float)dmin + 0.5f * (float)(dmax - dmin - 1);
  const int n4 = (kC * kH * kW) >> 2;
  float4 v = make_float4(val, val, val, val);
  float4* o = reinterpret_cast<float4*>(out);
  for (int i = blockIdx.x * blockDim.x + threadIdx.x; i < n4;
       i += gridDim.x * blockDim.x)
    o[i] = v;
}

// Channels [C, 2C): one block per (c,h) row of feaR.
// Softargmax over the 64-deep backward window, numerator+denominator reduced
// on the matrix pipe with a V_WMMA_F32_16X16X4_F32 accumulation chain
// (B col 0 = ones -> denominator, B col 1 = disparity -> numerator).
__global__ void __launch_bounds__(256)
gcnet_softargmax(const float* __restrict__ feaR, float* __restrict__ out,
                 const int* __restrict__ min_disp,
                 const int* __restrict__ max_disp) {
  __shared__ float row[kPad + kW];
  __shared__ float denb[kW];
  __shared__ float numb[kW];

  // force scalar (SGPR) values so the fast-path branch is s_cbranch,
  // keeping EXEC all-ones around the WMMA chain
  const int dmin = __builtin_amdgcn_readfirstlane(min_disp[0] >> 1);
  const int dmax = __builtin_amdgcn_readfirstlane(max_disp[0] >> 1);
  const int D = dmax - dmin;
  const float dminf = (float)dmin;

  const int t   = threadIdx.x;       // == w, 0..255
  const int blk = blockIdx.x;        // c*kH + h

  // ---- stage row (b128 loads) + zero pad ----
  if (t < (kW / 4)) {
    const float4* src4 = reinterpret_cast<const float4*>(feaR + (size_t)blk * kW);
    float4 v = src4[t];
    row[kPad + 4 * t + 0] = v.x;
    row[kPad + 4 * t + 1] = v.y;
    row[kPad + 4 * t + 2] = v.z;
    row[kPad + 4 * t + 3] = v.w;
  } else if (t < (kW / 4) + kPad) {
    row[t - kW / 4] = 0.0f;          // pad entries 0..kPad-1
  }
  __syncthreads();

  const int lane = t & 31;

  // ---- block-wide max (shift-invariance: one constant works for every
  // window; clamp to 0 for the implicit padding zeros). Redundant per wave,
  // so no extra barrier. ----
  float m = row[kPad + lane];
#pragma unroll
  for (int i = 1; i < 8; ++i) m = fmaxf(m, row[kPad + lane + 32 * i]);
#pragma unroll
  for (int k = 16; k >= 1; k >>= 1) m = fmaxf(m, __shfl_xor(m, k, 32));
  m = fmaxf(m, 0.0f);
  const float c0 = -m * kLog2e;      // e = exp2(v*log2e + c0)

  const size_t obase = (size_t)(kC * kH * kW) + (size_t)blk * kW;

  if (D == 64 && dmin == 0) {
    // ================= specialized fast path (reference setup) ===========
    const int w0   = t & ~31;        // wave covers w in [w0, w0+32)
    const int half = lane >> 4;      // A layout: lanes 0-15 K=0,1; 16-31 K=2,3
    const int nlo  = lane & 15;      // A row M / B column N
    const int kb   = half << 1;
    const float f0 = (nlo == 0) ? 1.0f : 0.0f;   // denominator column
    const float f1 = (nlo == 1) ? 1.0f : 0.0f;   // numerator (disparity) col
    const float binc = 4.0f * f1;

#pragma unroll
    for (int g = 0; g < 2; ++g) {    // two 16-row chains per wave
      const int wg   = w0 + (g << 4);
      const int wrow = wg + nlo;     // this lane's A-matrix row -> output w
      const float* pk = row + (kPad + wrow - kb);  // value(ch,j) = pk[-4ch-j]

      // batch all 32 window values into registers first (one dscnt wait)
      float vbuf[32];
#pragma unroll
      for (int ch = 0; ch < 16; ++ch) {
        vbuf[2 * ch]     = pk[-(4 * ch)];
        vbuf[2 * ch + 1] = pk[-(4 * ch) - 1];
      }

      v2f b;
      b[0] = f0 + f1 * (float)(kb + 0);
      b[1] = f0 + f1 * (float)(kb + 1);
      v8f acc = {0.f, 0.f, 0.f, 0.f, 0.f, 0.f, 0.f, 0.f};
#pragma unroll
      for (int ch = 0; ch < 16; ++ch) {
        v2f a;
        a[0] = __builtin_amdgcn_exp2f(__builtin_fmaf(vbuf[2 * ch],     kLog2e, c0));
        a[1] = __builtin_amdgcn_exp2f(__builtin_fmaf(vbuf[2 * ch + 1], kLog2e, c0));
        acc = __builtin_amdgcn_wmma_f32_16x16x4_f32(
            /*neg_a=*/false, a, /*neg_b=*/false, b,
            /*c_mod=*/(short)0, acc, /*reuse_a=*/false, /*reuse_b=*/false);
        b[0] += binc;                // next chunk's disparity values
        b[1] += binc;
      }

      // C/D layout: lane L, vgpr v -> D[v + 8*(L>=16)][L&15]
      if (nlo == 0) {                // N=0: denominators
#pragma unroll
        for (int v = 0; v < 8; ++v) denb[wg + v + (half << 3)] = acc[v];
      } else if (nlo == 1) {         // N=1: numerators
#pragma unroll
        for (int v = 0; v < 8; ++v) numb[wg + v + (half << 3)] = acc[v];
      }
    }
    // producers are lanes of this same wave; DScnt ordering suffices
    out[obase + t] = dminf + numb[t] / denb[t];
  } else {
    // ================= generic fallback (no wmma) ===========
    float den = 0.0f, num = 0.0f;
    for (int d = dmin; d < dmax; ++d) {
      int idx = kPad + t - d;
      idx = idx < 0 ? 0 : idx;       // clamp; value masked below
      float v = row[idx];
      v = (t - d >= 0) ? v : 0.0f;   // cost = 0 outside the image
      const float e = __builtin_amdgcn_exp2f(__builtin_fmaf(v, kLog2e, c0));
      den += e;
      num = __builtin_fmaf(e, (float)(d - dmin), num);
    }
    out[obase + t] = dminf + num / den;
  }
}

extern "C" void kernel_launch(void* const* d_in, const int* in_sizes, int n_in,
                              void* d_out, int out_size, void* d_ws,
                              size_t ws_size, hipStream_t stream) {
  // inputs: feaL (unused -- left half is analytically constant), feaR,
  //         min_disp, max_disp
  const float* feaR   = (const float*)d_in[1];
  const int* min_disp = (const int*)d_in[2];
  const int* max_disp = (const int*)d_in[3];
  float* out = (float*)d_out;

  gcnet_fill_left<<<1024, 256, 0, stream>>>(out, min_disp, max_disp);
  gcnet_softargmax<<<kC * kH, 256, 0, stream>>>(feaR, out, min_disp, max_disp);
}